// Loss_net_58110907515043
// MI455X (gfx1250) — compile-verified
//
#include <hip/hip_runtime.h>
#include <hip/hip_bf16.h>

// CDNA5 / gfx1250, wave32. f32 WMMA 16x16x4: A 16x4 (v2f/lane), B 4x16 (v2f/lane), C/D 16x16 (v8f/lane).
typedef float v2f __attribute__((ext_vector_type(2)));
typedef float v8f __attribute__((ext_vector_type(8)));

#define LOG2PI 1.8378770664093453f

__device__ __forceinline__ v8f wmma_f32_16x16x4(v2f a, v2f b) {
  v8f c = {};
  return __builtin_amdgcn_wmma_f32_16x16x4_f32(false, a, false, b, (short)0, c, false, false);
}

// Branch-free tanh: hardware V_TANH_F32 if available, else exp2/rcp identity.
__device__ __forceinline__ float fast_tanh(float x) {
#if __has_builtin(__builtin_amdgcn_tanhf)
  return __builtin_amdgcn_tanhf(x);
#else
  // tanh(x) = sign(x) * (1 - 2/(exp(2|x|)+1)); exp(2|x|) = exp2(2*log2(e)*|x|)
  float e = __builtin_amdgcn_exp2f(__builtin_fabsf(x) * 2.885390081777927f);
  float t = fmaf(-2.0f, __builtin_amdgcn_rcpf(e + 1.0f), 1.0f);
  return __builtin_copysignf(t, x);
#endif
}

__device__ __forceinline__ float wave_red(float v) {
#pragma unroll
  for (int o = 1; o < 32; o <<= 1) v += __shfl_xor(v, o, 32);
  return v;
}

// One ensemble evaluation: v(x,t) (and optionally div v) for 16 particles per wave.
// A[13]: W1 (lower half) / b1-in-K2 (upper half), bias trick. wt: phi-folded {W2_x, W2_y, what, 0} table.
template <bool WANT_DIV>
__device__ __forceinline__ void eval_ensemble(
    const v2f (&A)[13], const float4* __restrict__ wt, int half,
    float xe, float ye, float phA, float phB,
    float s00, float s01, float s10, float s11,
    float& vx, float& vy, float& dv) {
  v2f B;
  B.x = half ? 1.0f : xe;   // K=0 row: x ; K=2 row: 1.0 (bias)
  B.y = half ? 0.0f : ye;   // K=1 row: y ; K=3 row: 0
  float vxa = 0.f, vya = 0.f, dva = 0.f;
  const int ub = half ? 8 : 0;
#pragma unroll
  for (int t = 0; t < 13; ++t) {
    v8f acc = wmma_f32_16x16x4(A[t], B);   // preact = W1*x + b1, units 16t..16t+15
#pragma unroll
    for (int v = 0; v < 8; ++v) {
      float4 w = wt[16 * t + v + ub];      // half-broadcast ds_load
      float a = fast_tanh(acc[v]);
      vxa = fmaf(w.x, a, vxa);
      vya = fmaf(w.y, a, vya);
      if (WANT_DIV) dva = fmaf(w.z, fmaf(-a, a, 1.0f), dva);
    }
  }
  vxa += __shfl_xor(vxa, 16, 32);
  vya += __shfl_xor(vya, 16, 32);
  vx = fmaf(phA, s00, fmaf(phB, s10, vxa));
  vy = fmaf(phA, s01, fmaf(phB, s11, vya));
  if (WANT_DIV) { dva += __shfl_xor(dva, 16, 32); dv = dva; }
}

// ---------------- trajectory + loss1/loss2 kernel ----------------
__global__ __launch_bounds__(256) void traj_kernel(
    const float* __restrict__ x_in, const float* __restrict__ W1,
    const float* __restrict__ b1, const float* __restrict__ W2,
    const float* __restrict__ b2, float* __restrict__ ws) {
  __shared__ float4 wtab[9 * 208];  // per fi in {0..8}: {phi*W2x, phi*W2y, phi*what, 0}
  __shared__ float sb2[4];          // phi-unscaled b2 sums: [mblk*2 + k]
  __shared__ float rbuf[8][2];

  const int tid = threadIdx.x;
  const int lane = tid & 31;
  const int half = lane >> 4;
  const int wv = tid >> 5;
  const int p = (blockIdx.x * 8 + wv) * 16 + (lane & 15);

  float X = x_in[2 * p], Y = x_in[2 * p + 1];
  float lnR = -0.5f * (X * X + Y * Y) - LOG2PI;
  float l1 = 0.f;

  for (int n = 0; n < 10; ++n) {
    __syncthreads();
    if (tid < 208) {
      float w20 = 0.f, w21 = 0.f, w10 = 0.f, w11 = 0.f;
      if (tid < 200) {
        int g = n * 5 + tid / 20, hh = tid % 20;   // g = global net id (m*5+l)
        w20 = W2[g * 40 + hh];                     // W2[g,0,hh]
        w21 = W2[g * 40 + 20 + hh];                // W2[g,1,hh]
        w10 = W1[g * 40 + hh * 2];                 // W1[g,hh,0]
        w11 = W1[g * 40 + hh * 2 + 1];             // W1[g,hh,1]
      }
      float what = w20 * w10 + w21 * w11;          // trace term for divergence
      bool blk0 = (tid < 100);                     // m = n block vs m = n+1 block
#pragma unroll
      for (int fi = 0; fi < 9; ++fi) {
        float phB = fi * 0.125f;
        float ph = blk0 ? (1.0f - phB) : phB;
        wtab[fi * 208 + tid] = make_float4(ph * w20, ph * w21, ph * what, 0.f);
      }
    } else if (tid < 212) {
      int j = tid - 208, mb = j >> 1, kk = j & 1;
      float s = 0.f;
      for (int l = 0; l < 5; ++l) s += b2[(n * 5 + 5 * mb + l) * 2 + kk];
      sb2[j] = s;
    }
    __syncthreads();

    // A-operand: 13 tiles of 16 units; upper half carries b1 in the K=2 plane.
    v2f A[13];
#pragma unroll
    for (int t = 0; t < 13; ++t) {
      int ul = 16 * t + (lane & 15);
      float a0 = 0.f, a1 = 0.f;
      if (ul < 200) {
        int g = n * 5 + ul / 20, hh = ul % 20;
        if (half == 0) { a0 = W1[g * 40 + hh * 2]; a1 = W1[g * 40 + hh * 2 + 1]; }
        else           { a0 = b1[g * 20 + hh]; }
      }
      A[t].x = a0; A[t].y = a1;
    }
    float s00 = sb2[0], s01 = sb2[1], s10 = sb2[2], s11 = sb2[3];

    float dacc = 0.f;
    for (int s = 0; s < 4; ++s) {  // four RK4(step=h/4) sub-stages
      float phB0 = (2 * s) * 0.125f, phB1 = (2 * s + 1) * 0.125f, phB2 = (2 * s + 2) * 0.125f;
      const float4* r0 = &wtab[(2 * s) * 208];
      const float4* r1 = &wtab[(2 * s + 1) * 208];
      const float4* r2 = &wtab[(2 * s + 2) * 208];
      float k1x, k1y, k2x, k2y, k3x, k3y, k4x, k4y, dd, du;
      eval_ensemble<true>(A, r0, half, X, Y, 1.f - phB0, phB0, s00, s01, s10, s11, k1x, k1y, dd);
      float cf = (s == 1 || s == 3) ? 4.f : ((s == 2) ? 2.f : 1.f);  // d1+4d2+2d3+4d2'
      dacc = fmaf(cf, dd, dacc);
      float lw = (s == 0) ? 1.f : ((s == 2) ? 4.f : 0.f);
      l1 = fmaf(lw, fmaf(k1x, k1x, k1y * k1y), l1);
      eval_ensemble<false>(A, r1, half, fmaf(0.0125f, k1x, X), fmaf(0.0125f, k1y, Y),
                           1.f - phB1, phB1, s00, s01, s10, s11, k2x, k2y, du);
      eval_ensemble<false>(A, r1, half, fmaf(0.0125f, k2x, X), fmaf(0.0125f, k2y, Y),
                           1.f - phB1, phB1, s00, s01, s10, s11, k3x, k3y, du);
      eval_ensemble<false>(A, r2, half, fmaf(0.025f, k3x, X), fmaf(0.025f, k3y, Y),
                           1.f - phB2, phB2, s00, s01, s10, s11, k4x, k4y, du);
      X = fmaf(0.004166666883f, k1x + 2.f * (k2x + k3x) + k4x, X);  // h/24
      Y = fmaf(0.004166666883f, k1y + 2.f * (k2y + k3y) + k4y, Y);
    }
    // final eval at t_n + h : loss1 endpoint and d3'
    float vfx, vfy, ddf;
    eval_ensemble<true>(A, &wtab[8 * 208], half, X, Y, 0.f, 1.f, s00, s01, s10, s11, vfx, vfy, ddf);
    l1 += fmaf(vfx, vfx, vfy * vfy);
    lnR = fmaf(-0.008333333767f, dacc + ddf, lnR);  // -(h/12)*(d1+4d2+2d3+4d2'+d3')
  }

  float t2 = lnR + 0.5f * ((X - 4.f) * (X - 4.f) + (Y - 4.f) * (Y - 4.f)) + LOG2PI;
  float r1 = wave_red(l1);
  float r2 = wave_red(t2);
  if (lane == 0) { rbuf[wv][0] = r1; rbuf[wv][1] = r2; }
  __syncthreads();
  if (tid == 0) {
    float S1 = 0.f, S2 = 0.f;
    for (int w = 0; w < 8; ++w) { S1 += rbuf[w][0]; S2 += rbuf[w][1]; }
    ws[blockIdx.x] = S1;        // loss1 raw (double-counted across half-waves)
    ws[128 + blockIdx.x] = S2;  // loss2 raw
  }
}

// ---------------- gradV penalty kernel ----------------
__global__ __launch_bounds__(256) void gradv_kernel(
    const float* __restrict__ Xu, const float* __restrict__ W1,
    const float* __restrict__ b1, const float* __restrict__ W2,
    float* __restrict__ ws) {
  __shared__ float4 U4[112];  // {W2x*W1x, W2x*W1y, W2y*W1x, W2y*W1y}
  __shared__ float rbuf[8];
  const int tid = threadIdx.x, lane = tid & 31, half = lane >> 4, wv = tid >> 5;
  const int p = (blockIdx.x * 8 + wv) * 16 + (lane & 15);
  float xs = fmaf(20.f, Xu[2 * p], -10.f);
  float ys = fmaf(20.f, Xu[2 * p + 1], -10.f);
  v2f B;
  B.x = half ? 1.0f : xs;
  B.y = half ? 0.0f : ys;
  float gacc = 0.f;

  for (int k = 0; k <= 10; ++k) {  // Phi peaks: only m==k active, weight 1
    __syncthreads();
    if (tid < 112) {
      float4 u4 = make_float4(0.f, 0.f, 0.f, 0.f);
      if (tid < 100) {
        int g = k * 5 + tid / 20, hh = tid % 20;
        float w20 = W2[g * 40 + hh], w21 = W2[g * 40 + 20 + hh];
        float w10 = W1[g * 40 + hh * 2], w11 = W1[g * 40 + hh * 2 + 1];
        u4 = make_float4(w20 * w10, w20 * w11, w21 * w10, w21 * w11);
      }
      U4[tid] = u4;
    }
    __syncthreads();
    v2f A[7];
#pragma unroll
    for (int t = 0; t < 7; ++t) {
      int ul = 16 * t + (lane & 15);
      float a0 = 0.f, a1 = 0.f;
      if (ul < 100) {
        int g = k * 5 + ul / 20, hh = ul % 20;
        if (half == 0) { a0 = W1[g * 40 + hh * 2]; a1 = W1[g * 40 + hh * 2 + 1]; }
        else           { a0 = b1[g * 20 + hh]; }
      }
      A[t].x = a0; A[t].y = a1;
    }
    float J00 = 0.f, J01 = 0.f, J10 = 0.f, J11 = 0.f;
    const int ub = half ? 8 : 0;
#pragma unroll
    for (int t = 0; t < 7; ++t) {
      v8f acc = wmma_f32_16x16x4(A[t], B);
#pragma unroll
      for (int v = 0; v < 8; ++v) {
        float4 u4 = U4[16 * t + v + ub];
        float a = fast_tanh(acc[v]);
        float c = fmaf(-a, a, 1.0f);  // 1 - tanh^2
        J00 = fmaf(c, u4.x, J00); J01 = fmaf(c, u4.y, J01);
        J10 = fmaf(c, u4.z, J10); J11 = fmaf(c, u4.w, J11);
      }
    }
    J00 += __shfl_xor(J00, 16, 32); J01 += __shfl_xor(J01, 16, 32);
    J10 += __shfl_xor(J10, 16, 32); J11 += __shfl_xor(J11, 16, 32);
    float gk = J00 * J00 + J01 * J01 + J10 * J10 + J11 * J11;
    float wk = (k == 0 || k == 10) ? 1.f : 2.f;  // trapezoid weight
    gacc = fmaf(wk, gk, gacc);
  }
  float rg = wave_red(gacc);
  if (lane == 0) rbuf[wv] = rg;
  __syncthreads();
  if (tid == 0) {
    float S = 0.f;
    for (int w = 0; w < 8; ++w) S += rbuf[w];
    ws[256 + blockIdx.x] = S;
  }
}

// ---------------- deterministic final combine ----------------
__global__ void final_kernel(const float* __restrict__ ws, float* __restrict__ out) {
  if (threadIdx.x == 0 && blockIdx.x == 0) {
    float S1 = 0.f, S2 = 0.f, SG = 0.f;
    for (int b = 0; b < 128; ++b) { S1 += ws[b]; S2 += ws[128 + b]; SG += ws[256 + b]; }
    // 0.5f undoes the half-wave double counting of per-particle terms.
    float loss1 = 0.5f * S1 * (0.1f / (6.0f * 16384.0f));  // h/(6r)
    float loss2 = 0.5f * S2 * (1.0f / 16384.0f);           // lam * mean
    float loss3 = 0.1f * (0.5f * SG * (1.0f / 20.0f));     // alpha * Int, Int = sum/(2N)
    out[0] = loss1 + loss2 + loss3;
    out[1] = loss1;
    out[2] = loss2;
    out[3] = loss3;
  }
}

extern "C" void kernel_launch(void* const* d_in, const int* in_sizes, int n_in,
                              void* d_out, int out_size, void* d_ws, size_t ws_size,
                              hipStream_t stream) {
  const float* x  = (const float*)d_in[0];
  const float* Xu = (const float*)d_in[1];
  const float* W1 = (const float*)d_in[2];
  const float* b1 = (const float*)d_in[3];
  const float* W2 = (const float*)d_in[4];
  const float* b2 = (const float*)d_in[5];
  float* ws = (float*)d_ws;
  traj_kernel<<<128, 256, 0, stream>>>(x, W1, b1, W2, b2, ws);
  gradv_kernel<<<128, 256, 0, stream>>>(Xu, W1, b1, W2, ws);
  final_kernel<<<1, 32, 0, stream>>>(ws, (float*)d_out);
}